// HeteroGCNConv_14147622273722
// MI455X (gfx1250) — compile-verified
//
#include <hip/hip_runtime.h>
#include <hip/hip_bf16.h>

#define BLOCK   256
#define WAVES   (BLOCK / 32)
#define AG_ROWS 64            // dst rows per block in the aggregate kernel
#define AG_CAP  3072          // staged CSR entries cap (3072 * 8B = 24KB LDS)

// ---- gfx1250 async global->LDS copy (ASYNCcnt-tracked) ---------------------
__device__ __forceinline__ void async_ld_b64(unsigned lds_addr, unsigned long long gaddr) {
    asm volatile("global_load_async_to_lds_b64 %0, %1, off"
                 :: "v"(lds_addr), "v"(gaddr)
                 : "memory");
}
__device__ __forceinline__ void wait_async0() {
    asm volatile("s_wait_asynccnt 0" ::: "memory");
}
__device__ __forceinline__ int rfl(int x) { return __builtin_amdgcn_readfirstlane(x); }

// ---- 1) zero deg_src, deg_dst, cursor --------------------------------------
__global__ void k_init(unsigned* __restrict__ deg_src, unsigned* __restrict__ deg_dst,
                       unsigned* __restrict__ cursor, int n_src, int n_dst) {
    const int total = n_src + 2 * n_dst;
    for (int i = blockIdx.x * BLOCK + threadIdx.x; i < total; i += gridDim.x * BLOCK) {
        if (i < n_src)               deg_src[i] = 0u;
        else if (i < n_src + n_dst)  deg_dst[i - n_src] = 0u;
        else                         cursor[i - n_src - n_dst] = 0u;
    }
}

// ---- 2) out/in degrees via u32 atomics -------------------------------------
__global__ void k_degree(const long long* __restrict__ es, const long long* __restrict__ ed,
                         unsigned* __restrict__ deg_src, unsigned* __restrict__ deg_dst,
                         int n_edges) {
    for (int i = blockIdx.x * BLOCK + threadIdx.x; i < n_edges; i += gridDim.x * BLOCK) {
        atomicAdd(&deg_src[(int)es[i]], 1u);
        atomicAdd(&deg_dst[(int)ed[i]], 1u);
    }
}

// ---- 3) norm = clip(deg,1)^-0.5 --------------------------------------------
__global__ void k_norm(const unsigned* __restrict__ deg_src, const unsigned* __restrict__ deg_dst,
                       float* __restrict__ src_norm, float* __restrict__ dst_norm,
                       int n_src, int n_dst) {
    const int n = n_src > n_dst ? n_src : n_dst;
    for (int i = blockIdx.x * BLOCK + threadIdx.x; i < n; i += gridDim.x * BLOCK) {
        if (i < n_src) src_norm[i] = rsqrtf(fmaxf((float)deg_src[i], 1.0f));
        if (i < n_dst) dst_norm[i] = rsqrtf(fmaxf((float)deg_dst[i], 1.0f));
    }
}

// ---- 4a) per-block exclusive scan of deg_dst -> offs, block totals ---------
__global__ void k_scan_part(const unsigned* __restrict__ deg, unsigned* __restrict__ offs,
                            unsigned* __restrict__ partials, int n) {
    __shared__ unsigned sh[BLOCK];
    const int i = blockIdx.x * BLOCK + threadIdx.x;
    const unsigned v = (i < n) ? deg[i] : 0u;
    sh[threadIdx.x] = v;
    __syncthreads();
    for (int ofs = 1; ofs < BLOCK; ofs <<= 1) {
        unsigned t = (threadIdx.x >= (unsigned)ofs) ? sh[threadIdx.x - ofs] : 0u;
        __syncthreads();
        sh[threadIdx.x] += t;
        __syncthreads();
    }
    if (i < n) offs[i] = sh[threadIdx.x] - v;                 // exclusive in-block
    if (threadIdx.x == BLOCK - 1) partials[blockIdx.x] = sh[BLOCK - 1];
}

// ---- 4b) single-block chunked exclusive scan of partials (with carry) ------
__global__ void k_scan_tot(unsigned* __restrict__ partials, int nParts) {
    __shared__ unsigned sh[512];
    unsigned carry = 0;
    for (int base = 0; base < nParts; base += 512) {
        const int i = base + (int)threadIdx.x;
        const unsigned v = (i < nParts) ? partials[i] : 0u;
        sh[threadIdx.x] = v;
        __syncthreads();
        for (int ofs = 1; ofs < 512; ofs <<= 1) {
            unsigned t = (threadIdx.x >= (unsigned)ofs) ? sh[threadIdx.x - ofs] : 0u;
            __syncthreads();
            sh[threadIdx.x] += t;
            __syncthreads();
        }
        if (i < nParts) partials[i] = carry + sh[threadIdx.x] - v;   // exclusive + carry
        carry += sh[511];
        __syncthreads();
    }
}

// ---- 4c) add scanned block bases back --------------------------------------
__global__ void k_scan_add(unsigned* __restrict__ offs, const unsigned* __restrict__ partials,
                           int n) {
    const int i = blockIdx.x * BLOCK + threadIdx.x;
    if (i < n) offs[i] += partials[i >> 8];                   // BLOCK == 256
}

// ---- 5) build CSR: (src_idx, src_norm[src]) packed per edge ----------------
__global__ void k_build(const long long* __restrict__ es, const long long* __restrict__ ed,
                        const float* __restrict__ src_norm, const unsigned* __restrict__ offs,
                        unsigned* __restrict__ cursor, uint2* __restrict__ csr, int n_edges) {
    for (int i = blockIdx.x * BLOCK + threadIdx.x; i < n_edges; i += gridDim.x * BLOCK) {
        const int s = (int)es[i];
        const int d = (int)ed[i];
        const unsigned pos = offs[d] + atomicAdd(&cursor[d], 1u);
        csr[pos] = make_uint2((unsigned)s, __float_as_uint(src_norm[s]));
    }
}

// ---- helper: one gather + fma step -----------------------------------------
__device__ __forceinline__ void edge_fma(const float* __restrict__ src_emb, uint2 p,
                                         int lane, float2& acc) {
    const int   s = rfl((int)p.x);
    const float w = __uint_as_float((unsigned)rfl((int)p.y));
    const float2 v = ((const float2*)(src_emb + (size_t)s * 64))[lane];
    acc.x += v.x * w;
    acc.y += v.y * w;
}

// ---- 6) pull aggregation: wave per dst row, async-staged CSR slice ---------
__global__ void k_aggregate(const float* __restrict__ src_emb, const uint2* __restrict__ csr,
                            const unsigned* __restrict__ offs, const unsigned* __restrict__ deg,
                            const float* __restrict__ dst_norm, float* __restrict__ out,
                            int n_dst) {
    __shared__ uint2 sh[AG_CAP];
    const int r0 = blockIdx.x * AG_ROWS;
    if (r0 >= n_dst) return;
    const int rEnd = min(r0 + AG_ROWS, n_dst);

    const unsigned start = offs[r0];
    const unsigned end   = offs[rEnd - 1] + deg[rEnd - 1];   // block's CSR range is contiguous
    const unsigned cnt   = end - start;
    const bool staged    = (cnt <= AG_CAP);

    if (staged) {
        for (unsigned k = threadIdx.x; k < cnt; k += BLOCK)
            async_ld_b64((unsigned)(unsigned long long)&sh[k],
                         (unsigned long long)&csr[start + k]);
        wait_async0();                 // each wave waits its own async copies
    }
    __syncthreads();                   // then everyone's copies are visible

    const int lane = threadIdx.x & 31;
    const int wav  = threadIdx.x >> 5;

    for (int r = r0 + wav; r < rEnd; r += WAVES) {
        const unsigned rb = offs[r];
        const unsigned b  = rb - start;         // LDS-relative base (if staged)
        const unsigned c  = deg[r];
        float2 acc = make_float2(0.f, 0.f);

        unsigned j = 0;
        for (; j + 3 < c; j += 4) {             // 4 independent gathers in flight
            const uint2 p0 = staged ? sh[b + j]     : csr[rb + j];
            const uint2 p1 = staged ? sh[b + j + 1] : csr[rb + j + 1];
            const uint2 p2 = staged ? sh[b + j + 2] : csr[rb + j + 2];
            const uint2 p3 = staged ? sh[b + j + 3] : csr[rb + j + 3];
            if (staged && j + 5 < c) {          // global_prefetch_b8, next group
                __builtin_prefetch(src_emb + (size_t)sh[b + j + 4].x * 64, 0, 1);
                __builtin_prefetch(src_emb + (size_t)sh[b + j + 5].x * 64, 0, 1);
            }
            edge_fma(src_emb, p0, lane, acc);
            edge_fma(src_emb, p1, lane, acc);
            edge_fma(src_emb, p2, lane, acc);
            edge_fma(src_emb, p3, lane, acc);
        }
        for (; j < c; ++j) {                    // tail
            const uint2 p = staged ? sh[b + j] : csr[rb + j];
            edge_fma(src_emb, p, lane, acc);
        }
        const float nrm = dst_norm[r];
        ((float2*)(out + (size_t)r * 64))[lane] = make_float2(acc.x * nrm, acc.y * nrm);
    }
}

extern "C" void kernel_launch(void* const* d_in, const int* in_sizes, int n_in,
                              void* d_out, int out_size, void* d_ws, size_t ws_size,
                              hipStream_t stream) {
    const float*     src_emb = (const float*)d_in[0];
    const long long* es      = (const long long*)d_in[1];   // int64 edge src
    const long long* ed      = (const long long*)d_in[2];   // int64 edge dst
    const int n_src   = in_sizes[0] / 64;
    const int n_edges = in_sizes[1];
    const int n_dst   = out_size / 64;
    float* out = (float*)d_out;

    const int nParts = (n_dst + BLOCK - 1) / BLOCK;

    // workspace layout (csr first: needs 8B alignment)
    uint2*    csr      = (uint2*)d_ws;
    unsigned* deg_src  = (unsigned*)(csr + n_edges);
    unsigned* deg_dst  = deg_src + n_src;
    unsigned* offs     = deg_dst + n_dst;
    unsigned* cursor   = offs + n_dst;
    unsigned* partials = cursor + n_dst;
    float*    src_norm = (float*)(partials + nParts);
    float*    dst_norm = src_norm + n_src;

    const int init_total = n_src + 2 * n_dst;
    k_init<<<(init_total + BLOCK - 1) / BLOCK, BLOCK, 0, stream>>>(deg_src, deg_dst, cursor,
                                                                   n_src, n_dst);

    int gb = (n_edges + BLOCK - 1) / BLOCK; if (gb > 4096) gb = 4096;
    k_degree<<<gb, BLOCK, 0, stream>>>(es, ed, deg_src, deg_dst, n_edges);

    const int nmax = n_src > n_dst ? n_src : n_dst;
    k_norm<<<(nmax + BLOCK - 1) / BLOCK, BLOCK, 0, stream>>>(deg_src, deg_dst,
                                                             src_norm, dst_norm, n_src, n_dst);

    k_scan_part<<<nParts, BLOCK, 0, stream>>>(deg_dst, offs, partials, n_dst);
    k_scan_tot<<<1, 512, 0, stream>>>(partials, nParts);
    k_scan_add<<<(n_dst + BLOCK - 1) / BLOCK, BLOCK, 0, stream>>>(offs, partials, n_dst);

    k_build<<<gb, BLOCK, 0, stream>>>(es, ed, src_norm, offs, cursor, csr, n_edges);

    const int agBlocks = (n_dst + AG_ROWS - 1) / AG_ROWS;
    k_aggregate<<<agBlocks, BLOCK, 0, stream>>>(src_emb, csr, offs, deg_dst,
                                                dst_norm, out, n_dst);
}